// Decoder_35433480192155
// MI455X (gfx1250) — compile-verified
//
#include <hip/hip_runtime.h>
#include <hip/hip_bf16.h>
#include <math.h>
#include <stdint.h>

typedef __attribute__((ext_vector_type(2))) float v2f;
typedef __attribute__((ext_vector_type(8))) float v8f;

// Problem constants (from reference)
#define BB 128
#define RR 36
#define VIS 2048
#define MM 1024
#define EE 512
#define ATT 512
#define DD 1024
#define VV 10000
#define TT 19   // max_len - 1

// ---------------------------------------------------------------------------
// CDNA5 async global->LDS copy (ASYNCcnt-tracked, bypasses VGPRs/loadcnt).
// lds_addr: 32-bit LDS byte address (low 32 bits of the generic pointer).
// ---------------------------------------------------------------------------
__device__ __forceinline__ void async_copy_b128(uint32_t lds_addr, const float* gaddr)
{
    asm volatile("global_load_async_to_lds_b128 %0, %1, off"
                 :: "v"(lds_addr), "v"(gaddr) : "memory");
}
__device__ __forceinline__ void wait_async0()
{
    asm volatile("s_wait_asynccnt 0x0" ::: "memory");
}
__device__ __forceinline__ uint32_t lds_addr_of(const void* p)
{
    // Generic LDS pointers are SHARED_BASE|offset; low 32 bits are the DS address.
    return (uint32_t)(uintptr_t)p;
}

// ---------------------------------------------------------------------------
// Generic fp32 WMMA GEMM: C(MxN) = A(MxK) @ B(NxK)^T  [+bias] [+C] [masked]
// Block: 256 threads = 8 waves, computes a 64x64 tile of C.
// Wave grid 4(M) x 2(N); each wave owns a 16x32 strip = two v8f accumulators,
// sharing one A fragment per v_wmma_f32_16x16x4_f32 pair.
// Global->LDS staging is double-buffered via global_load_async_to_lds_b128:
// tile k+1 streams in while tile k's 16 WMMAs run; s_wait_asynccnt 0 + barrier
// between chunks.
// Requirements: M % 64 == 0, K % 32 == 0 (true for all call sites); N guarded.
// flags: bit0 = accumulate into C, bit1 = add bias[col], bit2 = zero rows with t>=lens[row]
// ---------------------------------------------------------------------------
#define TM 64
#define TN 64
#define TK 32
#define LDS_STR 36   // 144B row stride: 16B-aligned for b128, bank-conflict-free frags

__global__ __launch_bounds__(256) void gemm_wmma_f32(
    float* __restrict__ C, int ldc,
    const float* __restrict__ A, int lda,
    const float* __restrict__ B, int ldb,
    const float* __restrict__ bias,
    const int* __restrict__ lens, int t,
    int M, int N, int K, int flags)
{
    __shared__ float As[2][TM * LDS_STR];
    __shared__ float Bs[2][TN * LDS_STR];

    const int tid   = threadIdx.x;
    const int mBase = blockIdx.y * TM;
    const int nBase = blockIdx.x * TN;
    const int lane  = tid & 31;
    const int wid   = tid >> 5;
    const int wm    = wid & 3;        // wave M index 0..3 (16 rows each)
    const int wn    = wid >> 2;       // wave N index 0..1 (32 cols each)
    const int l15   = lane & 15;
    const int lhalf = lane >> 4;      // 0: lanes 0-15, 1: lanes 16-31
    const int koff  = lhalf * 2;      // K sub-offset within 4-wide fragment

    // Per-thread loader coordinates: 2 b128 per thread per tile (512 float4/tile).
    const int lr0 = tid >> 3;               // rows 0..31
    const int lr1 = lr0 + 32;               // rows 32..63
    const int lc4 = (tid & 7) << 2;         // col offset 0,4,..,28

    auto load_tiles = [&](int p, int k0) {
        // A tile 64x32 (no row guard: M multiple of 64)
        async_copy_b128(lds_addr_of(&As[p][lr0 * LDS_STR + lc4]),
                        A + (size_t)(mBase + lr0) * lda + k0 + lc4);
        async_copy_b128(lds_addr_of(&As[p][lr1 * LDS_STR + lc4]),
                        A + (size_t)(mBase + lr1) * lda + k0 + lc4);
        // B tile 64x32 with N guard (zero-fill rows past N)
        if (nBase + lr0 < N) {
            async_copy_b128(lds_addr_of(&Bs[p][lr0 * LDS_STR + lc4]),
                            B + (size_t)(nBase + lr0) * ldb + k0 + lc4);
        } else {
            float* d = &Bs[p][lr0 * LDS_STR + lc4];
            d[0] = 0.f; d[1] = 0.f; d[2] = 0.f; d[3] = 0.f;
        }
        if (nBase + lr1 < N) {
            async_copy_b128(lds_addr_of(&Bs[p][lr1 * LDS_STR + lc4]),
                            B + (size_t)(nBase + lr1) * ldb + k0 + lc4);
        } else {
            float* d = &Bs[p][lr1 * LDS_STR + lc4];
            d[0] = 0.f; d[1] = 0.f; d[2] = 0.f; d[3] = 0.f;
        }
    };

    v8f acc0 = {0.f, 0.f, 0.f, 0.f, 0.f, 0.f, 0.f, 0.f};
    v8f acc1 = {0.f, 0.f, 0.f, 0.f, 0.f, 0.f, 0.f, 0.f};

    const int nChunks = K / TK;

    // Prologue: stage chunk 0 into buffer 0.
    load_tiles(0, 0);
    wait_async0();
    __syncthreads();

    for (int ch = 0; ch < nChunks; ++ch) {
        const int p = ch & 1;
        if (ch + 1 < nChunks)
            load_tiles(p ^ 1, (ch + 1) * TK);   // overlap with compute below

        // 8 K-steps x 2 WMMA on buffer p. A frag: M=wm*16+l15, K=kk+koff..+1
        // (lanes 0-15 hold K0/K1, lanes 16-31 K2/K3). B frags mirror, N across lanes.
        #pragma unroll
        for (int kk = 0; kk < TK; kk += 4) {
            v2f a  = *(const v2f*)&As[p][(wm * 16 + l15) * LDS_STR + kk + koff];
            v2f b0 = *(const v2f*)&Bs[p][(wn * 32 + l15) * LDS_STR + kk + koff];
            v2f b1 = *(const v2f*)&Bs[p][(wn * 32 + 16 + l15) * LDS_STR + kk + koff];
            acc0 = __builtin_amdgcn_wmma_f32_16x16x4_f32(
                false, a, false, b0, (short)0, acc0, false, false);
            acc1 = __builtin_amdgcn_wmma_f32_16x16x4_f32(
                false, a, false, b1, (short)0, acc1, false, false);
        }

        if (ch + 1 < nChunks) {
            wait_async0();      // next buffer fully landed in LDS
            __syncthreads();    // all waves done reading current buffer
        }
    }

    // Store: C/D layout — VGPR i holds (M = i + 8*lhalf, N = l15) of each 16x16 tile.
    const int col0 = nBase + wn * 32 + l15;
    const int col1 = col0 + 16;
    const float bv0 = ((flags & 2) && col0 < N) ? bias[col0] : 0.f;
    const float bv1 = ((flags & 2) && col1 < N) ? bias[col1] : 0.f;
    #pragma unroll
    for (int i = 0; i < 8; ++i) {
        const int row = mBase + wm * 16 + lhalf * 8 + i;
        const bool dead = (flags & 4) && (t >= lens[row]);
        if (col0 < N) {
            float v = acc0[i] + bv0;
            if (flags & 1) v += C[(size_t)row * ldc + col0];
            if (dead) v = 0.f;
            C[(size_t)row * ldc + col0] = v;
        }
        if (col1 < N) {
            float v = acc1[i] + bv1;
            if (flags & 1) v += C[(size_t)row * ldc + col1];
            if (dead) v = 0.f;
            C[(size_t)row * ldc + col1] = v;
        }
    }
}

// ---------------------------------------------------------------------------
// words[b][t][e] = embed_W[captions[b][t]][e]   (t < TT)
// ---------------------------------------------------------------------------
__global__ void embed_kernel(const int* __restrict__ cap,
                             const float* __restrict__ embW,
                             float* __restrict__ words)
{
    const int i = blockIdx.x * 256 + threadIdx.x;
    if (i >= BB * TT * EE) return;
    const int e = i & (EE - 1);
    const int t = (i >> 9) % TT;
    const int b = i / (TT * EE);
    const int w = cap[b * (TT + 1) + t];
    words[i] = embW[(size_t)w * EE + e];
}

// ---------------------------------------------------------------------------
// feas0[b][v] = mean_r visual[b][r][v]
// ---------------------------------------------------------------------------
__global__ void feas0_kernel(const float* __restrict__ visual,
                             float* __restrict__ feas0)
{
    const int i = blockIdx.x * 256 + threadIdx.x;   // BB*VIS threads
    const int b = i >> 11;
    const int v = i & (VIS - 1);
    float s = 0.f;
    #pragma unroll 4
    for (int r = 0; r < RR; ++r)
        s += visual[((size_t)b * RR + r) * VIS + v];
    feas0[i] = s * (1.f / (float)RR);
}

// ---------------------------------------------------------------------------
// scores[b][r] = sum_a relu(att_fea[b][r][a] + ah[b][a] + att_bias[r]) * Ww[a]
// One wave per (b, r); shuffle reduction.
// ---------------------------------------------------------------------------
__global__ void attscore_kernel(const float* __restrict__ att_fea,
                                const float* __restrict__ ah,
                                const float* __restrict__ att_bias,
                                const float* __restrict__ Ww,
                                float* __restrict__ scores)
{
    const int gw   = (blockIdx.x * 256 + threadIdx.x) >> 5;  // 0..BB*RR-1
    const int lane = threadIdx.x & 31;
    const int b = gw / RR, r = gw % RR;
    const float* af = att_fea + ((size_t)b * RR + r) * ATT;
    const float* hv = ah + (size_t)b * ATT;
    const float bias = att_bias[r];
    float s = 0.f;
    for (int a = lane; a < ATT; a += 32) {
        float v = af[a] + hv[a] + bias;
        v = v > 0.f ? v : 0.f;
        s += v * Ww[a];
    }
    for (int m = 16; m; m >>= 1) s += __shfl_xor(s, m, 32);
    if (lane == 0) scores[gw] = s;
}

// ---------------------------------------------------------------------------
// Softmax over R + context: feas[b][v] = sum_r alpha[b][r] * visual[b][r][v]
// (t==0: feas = feas0). One block per batch row.
// ---------------------------------------------------------------------------
__global__ void ctx_kernel(const float* __restrict__ scores,
                           const float* __restrict__ visual,
                           const float* __restrict__ feas0,
                           float* __restrict__ feas, int t)
{
    __shared__ float sAl[RR];
    const int b = blockIdx.x;
    const int tid = threadIdx.x;
    if (t == 0) {
        for (int v = tid; v < VIS; v += 256)
            feas[(size_t)b * VIS + v] = feas0[(size_t)b * VIS + v];
        return;
    }
    if (tid == 0) {
        float mx = -1e30f;
        for (int r = 0; r < RR; ++r) mx = fmaxf(mx, scores[b * RR + r]);
        float sum = 0.f;
        for (int r = 0; r < RR; ++r) {
            float e = __expf(scores[b * RR + r] - mx);
            sAl[r] = e; sum += e;
        }
        const float inv = 1.f / sum;
        for (int r = 0; r < RR; ++r) sAl[r] *= inv;
    }
    __syncthreads();
    for (int v = tid; v < VIS; v += 256) {
        float s = 0.f;
        #pragma unroll 4
        for (int r = 0; r < RR; ++r)
            s += sAl[r] * visual[((size_t)b * RR + r) * VIS + v];
        feas[(size_t)b * VIS + v] = s;
    }
}

// ---------------------------------------------------------------------------
// LSTM pointwise: gates order [i | f | g | o] along 4*D; masked in-place h/c.
// ---------------------------------------------------------------------------
__global__ void lstm_kernel(const float* __restrict__ gates,
                            const float* __restrict__ bih,
                            const float* __restrict__ bhh,
                            const int* __restrict__ lens,
                            float* __restrict__ h, float* __restrict__ c, int t)
{
    const int i = blockIdx.x * 256 + threadIdx.x;   // BB*DD threads
    const int b = i >> 10;
    const int d = i & (DD - 1);
    const float* g = gates + (size_t)b * 4 * DD;
    const float gi = g[d]          + bih[d]          + bhh[d];
    const float gf = g[DD + d]     + bih[DD + d]     + bhh[DD + d];
    const float gg = g[2 * DD + d] + bih[2 * DD + d] + bhh[2 * DD + d];
    const float go = g[3 * DD + d] + bih[3 * DD + d] + bhh[3 * DD + d];
    const float si = 1.f / (1.f + __expf(-gi));
    const float sf = 1.f / (1.f + __expf(-gf));
    const float so = 1.f / (1.f + __expf(-go));
    const float cn = sf * c[i] + si * tanhf(gg);
    const float hn = so * tanhf(cn);
    if (t < lens[b]) { c[i] = cn; h[i] = hn; }
}

// ---------------------------------------------------------------------------
extern "C" void kernel_launch(void* const* d_in, const int* in_sizes, int n_in,
                              void* d_out, int out_size, void* d_ws, size_t ws_size,
                              hipStream_t stream)
{
    (void)in_sizes; (void)n_in; (void)out_size; (void)ws_size;
    const float* visual   = (const float*)d_in[0];
    const float* joint    = (const float*)d_in[1];
    const int*   captions = (const int*)d_in[2];
    const int*   lengths  = (const int*)d_in[3];
    /* d_in[4] = max_len (compile-time 20) */
    const float* embW     = (const float*)d_in[5];
    const float* Wih      = (const float*)d_in[6];
    const float* bih      = (const float*)d_in[7];
    const float* Whh      = (const float*)d_in[8];
    const float* bhh      = (const float*)d_in[9];
    const float* W_init_h = (const float*)d_in[10];
    const float* b_init_h = (const float*)d_in[11];
    const float* W_init_c = (const float*)d_in[12];
    const float* b_init_c = (const float*)d_in[13];
    const float* Wv       = (const float*)d_in[14];
    const float* Wh       = (const float*)d_in[15];
    const float* att_bias = (const float*)d_in[16];
    const float* Ww       = (const float*)d_in[17];
    const float* Wout     = (const float*)d_in[18];
    const float* bout     = (const float*)d_in[19];
    float* out = (float*)d_out;
    float* ws  = (float*)d_ws;

    // Workspace layout (floats)
    float* words   = ws;                                    // BB*TT*EE
    float* h       = words   + (size_t)BB * TT * EE;        // BB*DD
    float* c       = h       + (size_t)BB * DD;             // BB*DD
    float* feas0   = c       + (size_t)BB * DD;             // BB*VIS
    float* att_fea = feas0   + (size_t)BB * VIS;            // BB*RR*ATT
    float* ah      = att_fea + (size_t)BB * RR * ATT;       // BB*ATT
    float* scores  = ah      + (size_t)BB * ATT;            // BB*RR
    float* feas    = scores  + (size_t)BB * RR;             // BB*VIS
    float* gates   = feas    + (size_t)BB * VIS;            // BB*4*DD

    const dim3 blk(256);
    auto gemm = [&](float* C, int ldc, const float* A, int lda,
                    const float* Bm, int ldb, const float* bias,
                    const int* lens, int t, int M, int N, int K, int flags) {
        dim3 grid((N + TN - 1) / TN, M / TM);
        gemm_wmma_f32<<<grid, blk, 0, stream>>>(C, ldc, A, lda, Bm, ldb,
                                                bias, lens, t, M, N, K, flags);
    };

    // ---- One-time prologue ----
    embed_kernel<<<(BB * TT * EE + 255) / 256, blk, 0, stream>>>(captions, embW, words);
    // h0 = joint @ W_init_h^T + b_init_h ; c0 likewise
    gemm(h, DD, joint, MM, W_init_h, MM, b_init_h, nullptr, 0, BB, DD, MM, 2);
    gemm(c, DD, joint, MM, W_init_c, MM, b_init_c, nullptr, 0, BB, DD, MM, 2);
    feas0_kernel<<<(BB * VIS) / 256, blk, 0, stream>>>(visual, feas0);
    // att_fea = visual @ Wv^T   (4608 x 512, K=2048)
    gemm(att_fea, ATT, visual, VIS, Wv, VIS, nullptr, nullptr, 0, BB * RR, ATT, VIS, 0);

    // ---- Time loop ----
    for (int t = 0; t < TT; ++t) {
        if (t > 0) {
            // ah = h @ Wh^T
            gemm(ah, ATT, h, DD, Wh, DD, nullptr, nullptr, 0, BB, ATT, DD, 0);
            attscore_kernel<<<(BB * RR) / 8, blk, 0, stream>>>(att_fea, ah, att_bias, Ww, scores);
        }
        ctx_kernel<<<BB, blk, 0, stream>>>(scores, visual, feas0, feas, t);

        // gates = feas @ Wih[:, :VIS]^T  + words_t @ Wih[:, VIS:]^T  + h @ Whh^T
        gemm(gates, 4 * DD, feas, VIS, Wih, VIS + EE, nullptr, nullptr, 0,
             BB, 4 * DD, VIS, 0);
        gemm(gates, 4 * DD, words + (size_t)t * EE, TT * EE, Wih + VIS, VIS + EE,
             nullptr, nullptr, 0, BB, 4 * DD, EE, 1);
        gemm(gates, 4 * DD, h, DD, Whh, DD, nullptr, nullptr, 0, BB, 4 * DD, DD, 1);

        lstm_kernel<<<(BB * DD) / 256, blk, 0, stream>>>(gates, bih, bhh, lengths, h, c, t);

        // out[:, t, :] = mask ? h @ Wout^T + bout : 0
        gemm(out + (size_t)t * VV, TT * VV, h, DD, Wout, DD, bout, lengths, t,
             BB, VV, DD, 2 | 4);
    }
}